// DSC_23252952941334
// MI455X (gfx1250) — compile-verified
//
#include <hip/hip_runtime.h>

typedef __attribute__((ext_vector_type(2))) float v2f;
typedef __attribute__((ext_vector_type(8))) float v8f;

#define B_   16384
#define L_   66
#define P_   128
#define MC_  128
#define H_   16
#define M_   32
#define NP_  (MC_ * P_)          // 16384 elements per (c,p) slice
#define KTOT 8192                // (L_-2) * P_
#define NSTEP (KTOT / 4)         // 2048 wmma K-steps
#define IL_  (H_ * (H_ + 1))     // 272

// ---------------------------------------------------------------------------
// Kernel 1: tiny coefficient precompute.
//   cj[j][i] = lambda[i] * varphi[j][i]                      (32 x 16)
//   ck[k][l] = sigma_ext[l] * phi_ext[k][l]                  (32 x 17)
//   E[d][i][l] = sum_{j+k=d} cj[j][i] * ck[k][l]             (63 x 16 x 17)
// ---------------------------------------------------------------------------
__global__ void coef_kernel(const float* __restrict__ sigma,
                            const float* __restrict__ phi,
                            const float* __restrict__ lambda_,
                            const float* __restrict__ varphi,
                            float* __restrict__ cj,
                            float* __restrict__ ck,
                            float* __restrict__ E) {
    __shared__ float s_cj[M_ * H_];
    __shared__ float s_ck[M_ * (H_ + 1)];
    const int tid = threadIdx.x;

    for (int t = tid; t < M_ * H_; t += blockDim.x) {
        int j = t >> 4, i = t & 15;
        s_cj[t] = lambda_[i] * varphi[j * H_ + i];
    }
    for (int t = tid; t < M_ * (H_ + 1); t += blockDim.x) {
        int k = t / (H_ + 1), l = t % (H_ + 1);
        float se = (l == 0) ? 1.0f : sigma[l - 1];
        float pe = (l == 0) ? 1.0f : phi[k * H_ + (l - 1)];
        s_ck[t] = se * pe;
    }
    __syncthreads();

    for (int t = tid; t < M_ * H_; t += blockDim.x)       cj[t] = s_cj[t];
    for (int t = tid; t < M_ * (H_ + 1); t += blockDim.x) ck[t] = s_ck[t];

    for (int e = tid; e < 63 * IL_; e += blockDim.x) {
        int d = e / IL_;
        int r = e % IL_;
        int i = r / (H_ + 1);
        int l = r % (H_ + 1);
        int jlo = d - (M_ - 1); if (jlo < 0) jlo = 0;
        int jhi = (d < M_ - 1) ? d : (M_ - 1);
        float acc = 0.0f;
        for (int j = jlo; j <= jhi; ++j)
            acc += s_cj[j * H_ + i] * s_ck[(d - j) * (H_ + 1) + l];
        E[e] = acc;
    }
}

// ---------------------------------------------------------------------------
// Kernel 2: build per-lag weight matrices W[d][c][p] (d = 0..63) and scatter
// them into the WMMA B-fragment layout:
//   GEMM k-index:  kk = (63-d)*128 + p   (Y column after skipping t=0,1)
//   step s = kk>>2, kr = kk&3
//   B 4x16 fp32 fragment: lane = (kr>>1)*16 + (c&15), vgpr = kr&1, ntile = c>>4
//   Bpack is stored so a lane reads its 2 B-VGPRs as one contiguous float2:
//   Bpack[ ((s*8 + ntile)*32 + lane)*2 + vgpr ]
// ---------------------------------------------------------------------------
__global__ __launch_bounds__(256)
void pack_kernel(const float* __restrict__ Mb0,
                 const float* __restrict__ Mb,
                 const float* __restrict__ M0l,
                 const float* __restrict__ Mil,
                 const float* __restrict__ cj,
                 const float* __restrict__ ck,
                 const float* __restrict__ E,
                 float* __restrict__ Bpack) {
    const int cp = blockIdx.x * blockDim.x + threadIdx.x;   // 0..16383
    const int c = cp >> 7;
    const int p = cp & 127;

    #pragma unroll
    for (int db = 0; db < 4; ++db) {
        float w[16];
        #pragma unroll
        for (int r = 0; r < 16; ++r) w[r] = 0.0f;

        if (db == 0) w[0] = Mb0[cp];                         // term 1 (lag 0)

        // term 2: d in [1,32]
        for (int i = 0; i < H_; ++i) {
            float v = Mb[i * NP_ + cp];
            #pragma unroll
            for (int r = 0; r < 16; ++r) {
                const int d = db * 16 + r;
                if (d >= 1 && d <= 32) w[r] += cj[(d - 1) * H_ + i] * v;
            }
        }
        // term 3: d in [1,32]
        for (int l = 0; l <= H_; ++l) {
            float v = M0l[l * NP_ + cp];
            #pragma unroll
            for (int r = 0; r < 16; ++r) {
                const int d = db * 16 + r;
                if (d >= 1 && d <= 32) w[r] += ck[(d - 1) * (H_ + 1) + l] * v;
            }
        }
        // term 4: d in [1,63]
        for (int il = 0; il < IL_; ++il) {
            float v = Mil[il * NP_ + cp];
            #pragma unroll
            for (int r = 0; r < 16; ++r) {
                const int d = db * 16 + r;
                if (d >= 1) w[r] += E[(d - 1) * IL_ + il] * v;
            }
        }
        // scatter into B-fragment layout
        #pragma unroll
        for (int r = 0; r < 16; ++r) {
            const int d  = db * 16 + r;
            const int kk = (63 - d) * P_ + p;
            const int s  = kk >> 2;
            const int kr = kk & 3;
            const int lane = (kr >> 1) * 16 + (c & 15);
            const int vg   = kr & 1;
            const int nt   = c >> 4;
            Bpack[(((s * 8) + nt) * 32 + lane) * 2 + vg] = w[r];
        }
    }
}

// ---------------------------------------------------------------------------
// Kernel 3: out(16384 x 128) = Y(16384 x 8192) * Wt(8192 x 128) with fp32 WMMA.
// One wave = one 16-row M-tile x 64-col N-half (4 accumulator tiles).
// A fragment (16x4 fp32): lane = (k>=2)*16 + m, vgpr = k&1 -> per-lane float2
// straight from y_hist row (8B aligned). B fragments from pre-swizzled Bpack.
// ---------------------------------------------------------------------------
__global__ __launch_bounds__(256)
void gemm_kernel(const float* __restrict__ Y,
                 const float* __restrict__ Bpack,
                 float* __restrict__ out) {
    const int lane  = threadIdx.x & 31;
    const int wave  = blockIdx.x * (blockDim.x >> 5) + (threadIdx.x >> 5);
    const int mtile = wave >> 1;
    const int nhalf = wave & 1;

    const int row = mtile * 16 + (lane & 15);
    const float* ap = Y + (size_t)row * (L_ * P_) + 2 * P_ + ((lane >> 4) * 2);
    const float* bp = Bpack + (size_t)(nhalf * 4) * 64 + lane * 2;

    v8f acc[4] = {};

    #pragma unroll 4
    for (int s = 0; s < NSTEP; ++s) {
        v2f a  = *(const v2f*)ap;
        v2f b0 = *(const v2f*)(bp + 0 * 64);
        v2f b1 = *(const v2f*)(bp + 1 * 64);
        v2f b2 = *(const v2f*)(bp + 2 * 64);
        v2f b3 = *(const v2f*)(bp + 3 * 64);
        acc[0] = __builtin_amdgcn_wmma_f32_16x16x4_f32(false, a, false, b0,
                                                       (short)0, acc[0], false, false);
        acc[1] = __builtin_amdgcn_wmma_f32_16x16x4_f32(false, a, false, b1,
                                                       (short)0, acc[1], false, false);
        acc[2] = __builtin_amdgcn_wmma_f32_16x16x4_f32(false, a, false, b2,
                                                       (short)0, acc[2], false, false);
        acc[3] = __builtin_amdgcn_wmma_f32_16x16x4_f32(false, a, false, b3,
                                                       (short)0, acc[3], false, false);
        ap += 4;
        bp += 512;
    }

    // C/D layout: vgpr r holds row (r + 8*(lane>>4)), col = lane&15 within tile
    const int rbase = mtile * 16 + 8 * (lane >> 4);
    const int cbase = nhalf * 64 + (lane & 15);
    #pragma unroll
    for (int t = 0; t < 4; ++t) {
        #pragma unroll
        for (int r = 0; r < 8; ++r) {
            out[(size_t)(rbase + r) * MC_ + cbase + t * 16] = acc[t][r];
        }
    }
}

// ---------------------------------------------------------------------------
extern "C" void kernel_launch(void* const* d_in, const int* in_sizes, int n_in,
                              void* d_out, int out_size, void* d_ws, size_t ws_size,
                              hipStream_t stream) {
    const float* y       = (const float*)d_in[0];   // (B, 66, 128)
    const float* Mb0     = (const float*)d_in[1];   // (128, 128)
    const float* Mb      = (const float*)d_in[2];   // (16, 128, 128)
    const float* M0l     = (const float*)d_in[3];   // (17, 128, 128)
    const float* Mil     = (const float*)d_in[4];   // (16, 17, 128, 128)
    const float* sigma   = (const float*)d_in[5];   // (16,)
    const float* phi     = (const float*)d_in[6];   // (32, 16)
    const float* lambda_ = (const float*)d_in[7];   // (16,)
    const float* varphi  = (const float*)d_in[8];   // (32, 16)
    float* out = (float*)d_out;                     // (B, 128)

    float* Bpack = (float*)d_ws;                    // NSTEP*8*64 = 1,048,576 floats (4 MB)
    float* cj    = Bpack + (size_t)NSTEP * 8 * 64;  // 512 floats
    float* ck    = cj + M_ * H_;                    // 544 floats
    float* E     = ck + M_ * (H_ + 1);              // 17,136 floats

    coef_kernel<<<1, 256, 0, stream>>>(sigma, phi, lambda_, varphi, cj, ck, E);
    pack_kernel<<<NP_ / 256, 256, 0, stream>>>(Mb0, Mb, M0l, Mil, cj, ck, E, Bpack);
    gemm_kernel<<<(B_ / 16) * 2 / 8, 256, 0, stream>>>(y, Bpack, out);
}